// OHEMLoss_71055938945250
// MI455X (gfx1250) — compile-verified
//
#include <hip/hip_runtime.h>
#include <math.h>

// ---------------------------------------------------------------------------
// OHEM loss for MI455X (gfx1250, wave32).
// Exact k-th order statistic via 3-pass radix select on bit-flipped f32 keys
// (11 + 11 + 10 bits), then fused log-softmax + masked mean.
// All inter-block accumulation is integer-atomic or fixed-tree -> deterministic.
// Final 32-lane reduction uses V_WMMA_F32_16X16X4_F32 (B = ones) per the
// documented A/D VGPR layouts.
// ---------------------------------------------------------------------------

#define NT       256
#define NB_HIST  1024
#define NB_LOSS  2048

// u32-word offsets into d_ws
#define WS_HIST0 0        // 2048 bins: key[31:21]
#define WS_HIST1 2048     // 2048 bins: key[20:10]
#define WS_HIST2 4096     // 1024 bins: key[9:0]
#define WS_POS   5120     // positive count
#define WS_B1    5121
#define WS_R1    5122
#define WS_B2    5123
#define WS_R2    5124
#define WS_FLAG  5125     // 1 = threshold branch active
#define WS_THR   5126     // threshold float bits
#define WS_PART  5632     // float partials: 2 * NB_LOSS

__device__ __forceinline__ unsigned flip_key(float f) {
  // order-preserving f32 -> u32 (ascending)
  unsigned u = __float_as_uint(f);
  unsigned m = (unsigned)(-(int)(u >> 31)) | 0x80000000u;
  return u ^ m;
}
__device__ __forceinline__ float unflip_key(unsigned k) {
  unsigned m = (k & 0x80000000u) ? 0x80000000u : 0xFFFFFFFFu;
  return __uint_as_float(k ^ m);
}

__global__ void k_zero(unsigned* __restrict__ ws, int n) {
  int i = blockIdx.x * blockDim.x + threadIdx.x;
  if (i < n) ws[i] = 0u;
}

// Pass 1: positive count + 2048-bin histogram of key[31:21] over negatives.
__global__ void k_hist0(const float2* __restrict__ pred2,
                        const int* __restrict__ label, int n,
                        unsigned* __restrict__ hist0,
                        unsigned* __restrict__ posCount) {
  __shared__ unsigned lh[4][2048];
  __shared__ unsigned pbuf[NT];
  const int tid = threadIdx.x;
  for (int j = tid; j < 4 * 2048; j += NT) ((unsigned*)lh)[j] = 0u;
  __syncthreads();
  const int rep = tid & 3;
  const int stride = gridDim.x * blockDim.x;
  unsigned pos = 0u;
  for (int i = blockIdx.x * blockDim.x + tid; i < n; i += stride) {
    __builtin_prefetch(&pred2[i + 4 * stride], 0, 0);
    __builtin_prefetch(&label[i + 4 * stride], 0, 0);
    int lab = label[i];
    float pv = pred2[i].y;            // C == 2: foreground score = pred[:,1]
    if (lab != 0) {
      pos++;
    } else {
      unsigned key = flip_key(pv);
      atomicAdd(&lh[rep][key >> 21], 1u);
    }
  }
  __syncthreads();
  for (int b = tid; b < 2048; b += NT) {
    unsigned s = lh[0][b] + lh[1][b] + lh[2][b] + lh[3][b];
    if (s) atomicAdd(&hist0[b], s);
  }
  pbuf[tid] = pos;
  __syncthreads();
  for (int off = NT / 2; off > 0; off >>= 1) {
    if (tid < off) pbuf[tid] += pbuf[tid + off];
    __syncthreads();
  }
  if (tid == 0 && pbuf[0]) atomicAdd(posCount, pbuf[0]);
}

// Refinement histograms. pass==1: key[20:10] among key>>21==b1 (2048 bins).
//                        pass==2: key[9:0] among key>>10==prefix22 (1024 bins).
__global__ void k_hist_refine(const float2* __restrict__ pred2,
                              const int* __restrict__ label, int n,
                              const unsigned* __restrict__ ws, int pass,
                              unsigned* __restrict__ hist) {
  __shared__ unsigned lh[4][2048];
  if (!ws[WS_FLAG]) return;  // uniform
  const int tid = threadIdx.x;
  for (int j = tid; j < 4 * 2048; j += NT) ((unsigned*)lh)[j] = 0u;
  __syncthreads();
  const unsigned prefix = (pass == 1) ? ws[WS_B1]
                                      : ((ws[WS_B1] << 11) | ws[WS_B2]);
  const int rep = tid & 3;
  const int stride = gridDim.x * blockDim.x;
  for (int i = blockIdx.x * blockDim.x + tid; i < n; i += stride) {
    __builtin_prefetch(&pred2[i + 4 * stride], 0, 0);
    __builtin_prefetch(&label[i + 4 * stride], 0, 0);
    int lab = label[i];
    float pv = pred2[i].y;
    if (lab == 0) {
      unsigned key = flip_key(pv);
      if (pass == 1) {
        if ((key >> 21) == prefix) atomicAdd(&lh[rep][(key >> 10) & 0x7FFu], 1u);
      } else {
        if ((key >> 10) == prefix) atomicAdd(&lh[rep][key & 0x3FFu], 1u);
      }
    }
  }
  __syncthreads();
  const int nbins = (pass == 1) ? 2048 : 1024;
  for (int b = tid; b < nbins; b += NT) {
    unsigned s = lh[0][b] + lh[1][b] + lh[2][b] + lh[3][b];
    if (s) atomicAdd(&hist[b], s);
  }
}

// Descending-rank scan of a histogram to locate the digit containing rank k.
__global__ void k_scan(unsigned* __restrict__ ws, int mode, int n) {
  __shared__ unsigned part[NT];
  const int tid = threadIdx.x;
  const unsigned* hist;
  int nbins;
  unsigned k;
  if (mode == 0) {
    unsigned pos = ws[WS_POS];
    unsigned negsum = 3u * pos;                 // OHEM ratio = 3
    unsigned numneg = (unsigned)n - pos;
    unsigned flag = (numneg > negsum && negsum > 0u) ? 1u : 0u;
    if (tid == 0) {
      ws[WS_FLAG] = flag;
      ws[WS_THR] = 0xFF800000u;                 // default -inf
    }
    if (!flag) return;                          // uniform exit (mask = all true)
    hist = ws + WS_HIST0; nbins = 2048; k = negsum;
  } else if (mode == 1) {
    if (!ws[WS_FLAG]) return;
    hist = ws + WS_HIST1; nbins = 2048; k = ws[WS_R1];
  } else {
    if (!ws[WS_FLAG]) return;
    hist = ws + WS_HIST2; nbins = 1024; k = ws[WS_R2];
  }
  const int bpt = nbins / NT;
  unsigned s = 0u;
  for (int j = 0; j < bpt; ++j) s += hist[tid * bpt + j];
  part[tid] = s;
  __syncthreads();
  if (tid == 0) {
    unsigned acc = 0u;
    int g = 0;
    for (int t = NT - 1; t >= 0; --t) {
      if (acc + part[t] >= k) { g = t; break; }
      acc += part[t];
    }
    int d = g * bpt;
    for (int b = g * bpt + bpt - 1; b >= g * bpt; --b) {
      if (acc + hist[b] >= k) { d = b; break; }
      acc += hist[b];
    }
    unsigned kRem = k - acc;
    if (mode == 0)      { ws[WS_B1] = (unsigned)d; ws[WS_R1] = kRem; }
    else if (mode == 1) { ws[WS_B2] = (unsigned)d; ws[WS_R2] = kRem; }
    else {
      unsigned key = (ws[WS_B1] << 21) | (ws[WS_B2] << 10) | (unsigned)d;
      ws[WS_THR] = __float_as_uint(unflip_key(key));
    }
  }
}

// Fused masked cross-entropy: per-block (sum, count) partials, fixed tree.
__global__ void k_loss(const float2* __restrict__ pred2,
                       const int* __restrict__ label, int n,
                       const unsigned* __restrict__ ws,
                       float* __restrict__ partials) {
  __shared__ float ssum[NT];
  __shared__ float scnt[NT];
  const int tid = threadIdx.x;
  const unsigned flag = ws[WS_FLAG];
  const float thr = __uint_as_float(ws[WS_THR]);
  const int stride = gridDim.x * blockDim.x;
  float s = 0.f, c = 0.f;
  for (int i = blockIdx.x * blockDim.x + tid; i < n; i += stride) {
    __builtin_prefetch(&pred2[i + 4 * stride], 0, 0);
    __builtin_prefetch(&label[i + 4 * stride], 0, 0);
    float2 p = pred2[i];
    int lab = label[i];
    bool is_pos = (lab != 0);
    bool m = flag ? ((p.y >= thr) || is_pos) : true;
    if (m) {
      float mx = fmaxf(p.x, p.y);
      float mn = fminf(p.x, p.y);
      float lse = mx + log1pf(expf(mn - mx));   // exact log-sum-exp, 2 classes
      float pick = (is_pos ? p.y : p.x) - lse;  // logp[label]
      s += pick;
      c += 1.f;
    }
  }
  ssum[tid] = s;
  scnt[tid] = c;
  __syncthreads();
  for (int off = NT / 2; off > 0; off >>= 1) {
    if (tid < off) { ssum[tid] += ssum[tid + off]; scnt[tid] += scnt[tid + off]; }
    __syncthreads();
  }
  if (tid == 0) {
    partials[2 * blockIdx.x]     = ssum[0];
    partials[2 * blockIdx.x + 1] = scnt[0];
  }
}

typedef float v2f __attribute__((ext_vector_type(2)));
typedef float v8f __attribute__((ext_vector_type(8)));

// Wave32-lane reduction with V_WMMA_F32_16X16X4_F32.
// A layout (16x4 f32): VGPR0 lanes0-15 = K0 (M=lane), lanes16-31 = K2 (M=lane-16);
// VGPR1 = K1/K3 (zeroed). With B = ones: D[m][n] = p[m] + p[m+16] for all n.
// D layout: lane0 VGPRs0-7 = rows 0-7 col 0; lane16 VGPRs0-7 = rows 8-15 col 0.
__device__ __forceinline__ float wave32_sum_wmma(float p, float* lds, int lane) {
#if __has_builtin(__builtin_amdgcn_wmma_f32_16x16x4_f32)
  v2f a; a[0] = p;   a[1] = 0.f;
  v2f b; b[0] = 1.f; b[1] = 1.f;
  v8f cacc = {};
  v8f d = __builtin_amdgcn_wmma_f32_16x16x4_f32(false, a, false, b,
                                                (short)0, cacc, false, false);
  float v = d[0] + d[1] + d[2] + d[3] + d[4] + d[5] + d[6] + d[7];
  lds[lane] = v;
  __syncthreads();
  return lds[0] + lds[16];
#else
  lds[lane] = p;
  __syncthreads();
  float t = 0.f;
  for (int i = 0; i < 32; ++i) t += lds[i];
  return t;
#endif
}

__global__ void k_final(const float* __restrict__ partials, int nb,
                        float* __restrict__ out) {
  __shared__ float lS[32];
  __shared__ float lC[32];
  const int lane = threadIdx.x;  // single wave: 32 threads, EXEC all ones
  float s = 0.f, c = 0.f;
  for (int i = lane; i < nb; i += 32) {
    s += partials[2 * i];
    c += partials[2 * i + 1];
  }
  float S = wave32_sum_wmma(s, lS, lane);
  __syncthreads();
  float C = wave32_sum_wmma(c, lC, lane);
  if (lane == 0) out[0] = -S / C;
}

extern "C" void kernel_launch(void* const* d_in, const int* in_sizes, int n_in,
                              void* d_out, int out_size, void* d_ws, size_t ws_size,
                              hipStream_t stream) {
  const float2* pred2 = (const float2*)d_in[0];   // [N,2] f32, row-major
  const int* label    = (const int*)d_in[1];      // [N] i32
  const int n = in_sizes[1];
  unsigned* ws = (unsigned*)d_ws;
  float* partials = (float*)(ws + WS_PART);
  float* out = (float*)d_out;

  const int zeroN = WS_THR + 1;  // hist0/1/2 + counters + state must be zeroed
  k_zero<<<dim3((zeroN + NT - 1) / NT), dim3(NT), 0, stream>>>(ws, zeroN);
  k_hist0<<<dim3(NB_HIST), dim3(NT), 0, stream>>>(pred2, label, n,
                                                  ws + WS_HIST0, ws + WS_POS);
  k_scan<<<dim3(1), dim3(NT), 0, stream>>>(ws, 0, n);
  k_hist_refine<<<dim3(NB_HIST), dim3(NT), 0, stream>>>(pred2, label, n, ws, 1,
                                                        ws + WS_HIST1);
  k_scan<<<dim3(1), dim3(NT), 0, stream>>>(ws, 1, n);
  k_hist_refine<<<dim3(NB_HIST), dim3(NT), 0, stream>>>(pred2, label, n, ws, 2,
                                                        ws + WS_HIST2);
  k_scan<<<dim3(1), dim3(NT), 0, stream>>>(ws, 2, n);
  k_loss<<<dim3(NB_LOSS), dim3(NT), 0, stream>>>(pred2, label, n, ws, partials);
  k_final<<<dim3(1), dim3(32), 0, stream>>>(partials, NB_LOSS, out);
}